// StrictOrthogonal_10960756540132
// MI455X (gfx1250) — compile-verified
//
#include <hip/hip_runtime.h>
#include <math.h>

// ---------------------------------------------------------------------------
// StrictOrthogonal on MI455X (gfx1250, wave32)
//
// Reformulation: all inner products in the reference are quadratic forms in
// the 32x32 Gram matrix S = X^H X.  So:
//   Pass 1 (WMMA):  S = X^H X           (atomic f32 reduction, 8KB in L2)
//   Pass 2 (1 wave): exact coefficient-space replay of the sequential MGS
//                    recursion + 3 refinement sweeps -> 32x32 transform T
//   Pass 3 (WMMA):  Q = X * T           (complex GEMM, b64 stores)
// This replaces ~35 grid-wide sequential passes over HBM-sized vectors with
// two streaming passes over the 4MB input (L2-resident on the 192MB L2).
// ---------------------------------------------------------------------------

typedef float v2f __attribute__((ext_vector_type(2)));
typedef float v8f __attribute__((ext_vector_type(8)));

// D(16x16,f32) = A(16x4,f32) x B(4x16,f32) + C   -- v_wmma_f32_16x16x4_f32
__device__ __forceinline__ v8f wmma4(v2f a, v2f b, v8f c) {
  return __builtin_amdgcn_wmma_f32_16x16x4_f32(false, a, false, b,
                                               (short)0, c, false, false);
}

__device__ __forceinline__ float wave_sum32(float v) {
#pragma unroll
  for (int off = 16; off > 0; off >>= 1) v += __shfl_xor(v, off, 32);
  return v;
}

// ---------------------------------------------------------------------------
// Kernel 1: partial Gram matrices via WMMA, reduced with global f32 atomics.
//   Sr = Xr^T Xr + Xi^T Xi  (symmetric part of S = X^H X)
//   Si = Xr^T Xi - Xi^T Xr  (imag part; computed as RI - IR in VALU since
//                            f32 WMMA has no A/B negate per ISA 7.12 NEG table)
// A-fragment (16x4): lane l holds column (l&15) of X, K rows kb..kb+1 with
// kb = 2*(l>>4); the same fragment serves as the B operand (B 4x16: lane l
// holds N=(l&15), same K split), so one set of loads feeds both operands.
// ---------------------------------------------------------------------------
__global__ void __launch_bounds__(32)
so_gram_kernel(const float* __restrict__ x, float* __restrict__ Sr,
               float* __restrict__ Si, int rows_per_block) {
  const int lane = threadIdx.x;
  const int c    = lane & 15;
  const int kb   = (lane >> 4) << 1;
  const int m0   = blockIdx.x * rows_per_block;
  const float2* x2 = (const float2*)x;  // element (m,r): x2[m*32 + r] = (re,im)

  v8f s00 = {}, s01 = {}, s10 = {}, s11 = {};
  v8f p00 = {}, p01 = {}, p10 = {}, p11 = {};   // Xr^T Xi tiles
  v8f q00 = {}, q01 = {}, q10 = {}, q11 = {};   // Xi^T Xr tiles

  for (int m = m0; m < m0 + rows_per_block; m += 4) {
    const int r0 = m + kb;
    float2 f00 = x2[r0 * 32 + c];             // row r0,   cols 0..15
    float2 f01 = x2[(r0 + 1) * 32 + c];       // row r0+1, cols 0..15
    float2 f10 = x2[r0 * 32 + 16 + c];        // row r0,   cols 16..31
    float2 f11 = x2[(r0 + 1) * 32 + 16 + c];  // row r0+1, cols 16..31
    v2f Ar0 = {f00.x, f01.x}, Ai0 = {f00.y, f01.y};
    v2f Ar1 = {f10.x, f11.x}, Ai1 = {f10.y, f11.y};

    // Sr tiles: accumulate rr and ii products into the same accumulator
    s00 = wmma4(Ar0, Ar0, s00); s00 = wmma4(Ai0, Ai0, s00);
    s01 = wmma4(Ar0, Ar1, s01); s01 = wmma4(Ai0, Ai1, s01);
    s10 = wmma4(Ar1, Ar0, s10); s10 = wmma4(Ai1, Ai0, s10);
    s11 = wmma4(Ar1, Ar1, s11); s11 = wmma4(Ai1, Ai1, s11);
    // RI and IR products for Si
    p00 = wmma4(Ar0, Ai0, p00); q00 = wmma4(Ai0, Ar0, q00);
    p01 = wmma4(Ar0, Ai1, p01); q01 = wmma4(Ai0, Ar1, q01);
    p10 = wmma4(Ar1, Ai0, p10); q10 = wmma4(Ai1, Ar0, q10);
    p11 = wmma4(Ar1, Ai1, p11); q11 = wmma4(Ai1, Ar1, q11);
  }

  // C/D layout: VGPR v -> tile row v + 8*(lane>=16), tile col lane&15
  const int rsel = (lane >> 4) * 8;
#pragma unroll
  for (int v = 0; v < 8; ++v) {
    const int rr = rsel + v;
    atomicAdd(&Sr[rr * 32 + c],              s00[v]);
    atomicAdd(&Sr[rr * 32 + 16 + c],         s01[v]);
    atomicAdd(&Sr[(16 + rr) * 32 + c],       s10[v]);
    atomicAdd(&Sr[(16 + rr) * 32 + 16 + c],  s11[v]);
    atomicAdd(&Si[rr * 32 + c],              p00[v] - q00[v]);
    atomicAdd(&Si[rr * 32 + 16 + c],         p01[v] - q01[v]);
    atomicAdd(&Si[(16 + rr) * 32 + c],       p10[v] - q10[v]);
    atomicAdd(&Si[(16 + rr) * 32 + 16 + c],  p11[v] - q11[v]);
  }
}

// ---------------------------------------------------------------------------
// Kernel 2 (single wave32): coefficient-space replay.
// Phase A: sequential MGS with vectors represented as coefficient vectors
//   a wrt columns of X:  <u,v> = a_u^H S a_v (exact).  Maintain y = S a
//   incrementally (y -= c * S t_i), so each projection is one wave reduction.
// Phase B: G = T^H S T = T^H U, then 3 sweeps: since each refinement step
//   subtracts a multiple of q_i, only coefficient i changes ->
//   A[i][j] -= 0.5 * (G row i . A col j);  then normalize with b^H G b,
//   T <- T A, G <- A^H G A.
// ---------------------------------------------------------------------------
__global__ void __launch_bounds__(32)
so_solve_kernel(const float* __restrict__ SrG, const float* __restrict__ SiG,
                float* __restrict__ TrG, float* __restrict__ TiG) {
  __shared__ float Srs[32][32], Sis[32][32];
  __shared__ float Trs[32][32], Tis[32][32];
  __shared__ float Urs[32][32], Uis[32][32];  // U = S*T, reused as matmul temp
  __shared__ float Grs[32][32], Gis[32][32];
  __shared__ float Ars[32][32], Ais[32][32];
  const int k = threadIdx.x;  // lane = vector component / row / column index

  for (int r = 0; r < 32; ++r) {
    Srs[r][k] = SrG[r * 32 + k];
    Sis[r][k] = SiG[r * 32 + k];
  }
  __syncthreads();

  // --- Phase A: classical/modified GS pass, exact coefficient-space replay
  for (int j = 0; j < 32; ++j) {
    float a_re = (k == j) ? 1.0f : 0.0f, a_im = 0.0f;
    float y_re = Srs[k][j], y_im = Sis[k][j];  // y = S e_j
    for (int i = 0; i < j; ++i) {
      const float ur = Urs[k][i], ui = Uis[k][i];
      // c = (S t_i)^H a  (S Hermitian => equals t_i^H S a)
      const float cr = wave_sum32(ur * a_re + ui * a_im);
      const float ci = wave_sum32(ur * a_im - ui * a_re);
      const float tr = Trs[k][i], ti = Tis[k][i];
      a_re -= cr * tr - ci * ti;  a_im -= cr * ti + ci * tr;
      y_re -= cr * ur - ci * ui;  y_im -= cr * ui + ci * ur;
    }
    const float nrm = wave_sum32(a_re * y_re + a_im * y_im);  // Re(a^H S a)
    const float inv = 1.0f / sqrtf(nrm + 1e-10f);
    Trs[k][j] = a_re * inv;  Tis[k][j] = a_im * inv;
    Urs[k][j] = y_re * inv;  Uis[k][j] = y_im * inv;
    __syncthreads();
  }

  // --- G = T^H U  (lane k = row i of G)
  for (int jj = 0; jj < 32; ++jj) {
    float gr = 0.f, gi = 0.f;
    for (int kk = 0; kk < 32; ++kk) {
      const float tr = Trs[kk][k], ti = Tis[kk][k];
      const float ur = Urs[kk][jj], ui = Uis[kk][jj];
      gr += tr * ur + ti * ui;
      gi += tr * ui - ti * ur;
    }
    Grs[k][jj] = gr;  Gis[k][jj] = gi;
  }
  __syncthreads();

  // --- Phase B: 3 refinement sweeps (lane k owns column k of A)
  for (int s = 0; s < 3; ++s) {
    for (int r = 0; r < 32; ++r) {
      Ars[r][k] = (r == k) ? 1.0f : 0.0f;
      Ais[r][k] = 0.0f;
    }
    for (int i = 0; i < 32; ++i) {
      float cr = 0.f, ci = 0.f;
      for (int kk = 0; kk < 32; ++kk) {
        const float gr = Grs[i][kk], gi = Gis[i][kk];
        const float br = Ars[kk][k], bi = Ais[kk][k];
        cr += gr * br - gi * bi;
        ci += gr * bi + gi * br;
      }
      if (i != k) { Ars[i][k] -= 0.5f * cr; Ais[i][k] -= 0.5f * ci; }
    }
    // norm^2 = Re(b^H G b)
    float nrm = 0.f;
    for (int kk = 0; kk < 32; ++kk) {
      float yr = 0.f, yi = 0.f;
      for (int mm = 0; mm < 32; ++mm) {
        const float gr = Grs[kk][mm], gi = Gis[kk][mm];
        const float br = Ars[mm][k], bi = Ais[mm][k];
        yr += gr * br - gi * bi;
        yi += gr * bi + gi * br;
      }
      nrm += Ars[kk][k] * yr + Ais[kk][k] * yi;
    }
    const float inv = 1.0f / sqrtf(nrm + 1e-10f);
    for (int kk = 0; kk < 32; ++kk) { Ars[kk][k] *= inv; Ais[kk][k] *= inv; }
    __syncthreads();

    // W = T*A (into U), lane k = row
    for (int jj = 0; jj < 32; ++jj) {
      float wr = 0.f, wi = 0.f;
      for (int kk = 0; kk < 32; ++kk) {
        const float tr = Trs[k][kk], ti = Tis[k][kk];
        const float ar = Ars[kk][jj], ai = Ais[kk][jj];
        wr += tr * ar - ti * ai;
        wi += tr * ai + ti * ar;
      }
      Urs[k][jj] = wr;  Uis[k][jj] = wi;
    }
    __syncthreads();
    for (int jj = 0; jj < 32; ++jj) { Trs[k][jj] = Urs[k][jj]; Tis[k][jj] = Uis[k][jj]; }
    __syncthreads();

    // W = G*A (into U), lane k = row
    for (int jj = 0; jj < 32; ++jj) {
      float wr = 0.f, wi = 0.f;
      for (int kk = 0; kk < 32; ++kk) {
        const float gr = Grs[k][kk], gi = Gis[k][kk];
        const float ar = Ars[kk][jj], ai = Ais[kk][jj];
        wr += gr * ar - gi * ai;
        wi += gr * ai + gi * ar;
      }
      Urs[k][jj] = wr;  Uis[k][jj] = wi;
    }
    __syncthreads();
    // G = A^H W, lane k = row i
    for (int jj = 0; jj < 32; ++jj) {
      float gr = 0.f, gi = 0.f;
      for (int kk = 0; kk < 32; ++kk) {
        const float ar = Ars[kk][k], ai = Ais[kk][k];
        const float wr = Urs[kk][jj], wi = Uis[kk][jj];
        gr += ar * wr + ai * wi;
        gi += ar * wi - ai * wr;
      }
      Grs[k][jj] = gr;  Gis[k][jj] = gi;
    }
    __syncthreads();
  }

  for (int r = 0; r < 32; ++r) {
    TrG[r * 32 + k] = Trs[r][k];
    TiG[r * 32 + k] = Tis[r][k];
  }
}

// ---------------------------------------------------------------------------
// Kernel 3: Q = X * T  (M x 32 complex GEMM) via WMMA f32 16x16x4.
//   Qr = Xr*Tr - Xi*Ti  (subtraction via register-negated A fragment)
//   Qi = Xr*Ti + Xi*Tr
// ---------------------------------------------------------------------------
__global__ void __launch_bounds__(32)
so_apply_kernel(const float* __restrict__ x, const float* __restrict__ Tr,
                const float* __restrict__ Ti, float* __restrict__ out,
                int rows_per_block) {
  const int lane = threadIdx.x;
  const int c    = lane & 15;
  const int kb   = (lane >> 4) << 1;
  const int m0   = blockIdx.x * rows_per_block;

  // Preload all B fragments: B[k,n] = T[k0*4+k, n0+n]
  v2f Br[8][2], Bi[8][2];
#pragma unroll
  for (int k0 = 0; k0 < 8; ++k0) {
    const int kk = k0 * 4 + kb;
#pragma unroll
    for (int g = 0; g < 2; ++g) {
      const int n = g * 16 + c;
      v2f br = {Tr[kk * 32 + n], Tr[(kk + 1) * 32 + n]};
      v2f bi = {Ti[kk * 32 + n], Ti[(kk + 1) * 32 + n]};
      Br[k0][g] = br;
      Bi[k0][g] = bi;
    }
  }

  const float4* x4  = (const float4*)x;  // (re,im,re,im) for two adjacent cols
  float2* out2      = (float2*)out;
  const int rsel    = (lane >> 4) * 8;

  for (int m = m0; m < m0 + rows_per_block; m += 16) {
    v8f ar0 = {}, ai0 = {}, ar1 = {}, ai1 = {};
    const int r = m + c;  // A row handled by this lane
#pragma unroll
    for (int k0 = 0; k0 < 8; ++k0) {
      const int colx = k0 * 4 + kb;  // even -> float4-aligned
      float4 f = x4[r * 16 + (colx >> 1)];
      v2f Arf = {f.x, f.z};
      v2f Aif = {f.y, f.w};
      v2f Ain = {-f.y, -f.w};
      ar0 = wmma4(Arf, Br[k0][0], ar0); ar0 = wmma4(Ain, Bi[k0][0], ar0);
      ai0 = wmma4(Arf, Bi[k0][0], ai0); ai0 = wmma4(Aif, Br[k0][0], ai0);
      ar1 = wmma4(Arf, Br[k0][1], ar1); ar1 = wmma4(Ain, Bi[k0][1], ar1);
      ai1 = wmma4(Arf, Bi[k0][1], ai1); ai1 = wmma4(Aif, Br[k0][1], ai1);
    }
#pragma unroll
    for (int v = 0; v < 8; ++v) {
      const int row = m + rsel + v;
      out2[row * 32 + c]      = make_float2(ar0[v], ai0[v]);
      out2[row * 32 + 16 + c] = make_float2(ar1[v], ai1[v]);
    }
  }
}

// ---------------------------------------------------------------------------
extern "C" void kernel_launch(void* const* d_in, const int* in_sizes, int n_in,
                              void* d_out, int out_size, void* d_ws, size_t ws_size,
                              hipStream_t stream) {
  (void)n_in; (void)out_size; (void)ws_size;
  const float* x = (const float*)d_in[0];
  float* out = (float*)d_out;
  float* ws  = (float*)d_ws;
  float* Sr = ws;            // 32*32
  float* Si = ws + 1024;     // 32*32
  float* Tr = ws + 2048;     // 32*32
  float* Ti = ws + 3072;     // 32*32

  const int M = in_sizes[0] / 64;  // (M, 32, 2) floats
  hipMemsetAsync(ws, 0, 2048 * sizeof(float), stream);  // zero Gram accumulators

  const int ROWS = 128;            // rows of X per wave
  const int nb = M / ROWS;         // 16384/128 = 128 blocks (1 wave each)
  so_gram_kernel<<<nb, 32, 0, stream>>>(x, Sr, Si, ROWS);
  so_solve_kernel<<<1, 32, 0, stream>>>(Sr, Si, Tr, Ti);
  so_apply_kernel<<<nb, 32, 0, stream>>>(x, Tr, Ti, out, ROWS);
}